// DetectionLoss_7413113553027
// MI455X (gfx1250) — compile-verified
//
#include <hip/hip_runtime.h>
#include <hip/hip_bf16.h>

typedef __attribute__((ext_vector_type(2))) float v2f;
typedef __attribute__((ext_vector_type(8))) float v8f;

constexpr int N_TOT     = 130944;          // total anchors over 5 levels
constexpr int NUM_TILES = N_TOT / 16;      // 8184 WMMA column tiles
constexpr int NSEG      = 96;              // anchor-range segments for top-k pass
constexpr int TPS       = (NUM_TILES + NSEG - 1) / NSEG; // 86 tiles/segment
constexpr int TOPK      = 9;
constexpr int BQ        = 8;               // batch
constexpr int MGT       = 32;              // gt boxes per image
constexpr int LBLK      = 256;
constexpr int NBLK      = (N_TOT + LBLK - 1) / LBLK;     // 512
constexpr float EPSF    = 1e-7f;

struct Ptrs    { const float* cls[5]; const float* reg[5]; };
struct AncPtrs { const float* anc[5]; };

// ---------------- helpers ----------------

__device__ __forceinline__ float softplusf(float x) {
  // log(1 + e^x), stable
  return fmaxf(x, 0.0f) + log1pf(__expf(-fabsf(x)));
}

struct Heap { float d[TOPK]; int i[TOPK]; };

__device__ __forceinline__ void heap_init(Heap& h) {
#pragma unroll
  for (int k = 0; k < TOPK; ++k) { h.d[k] = 3.0e38f; h.i[k] = -1; }
}

// keep TOPK smallest, sorted ascending; fully unrolled -> stays in VGPRs
__device__ __forceinline__ void heap_insert(Heap& h, float v, int idx) {
  if (v < h.d[TOPK - 1]) {
    h.d[TOPK - 1] = v; h.i[TOPK - 1] = idx;
#pragma unroll
    for (int k = TOPK - 1; k > 0; --k) {
      if (h.d[k] < h.d[k - 1]) {
        float td = h.d[k]; h.d[k] = h.d[k - 1]; h.d[k - 1] = td;
        int   ti = h.i[k]; h.i[k] = h.i[k - 1]; h.i[k - 1] = ti;
      }
    }
  }
}

// anchor index -> (level, h, w, anchor, grid dim)
__device__ __forceinline__ void decomposeAnchor(int n, int& l, int& h, int& w,
                                                int& a, int& dim) {
  int off;
  if      (n <  98304) { l = 0; off = 0;      dim = 128; }
  else if (n < 122880) { l = 1; off = 98304;  dim = 64;  }
  else if (n < 129024) { l = 2; off = 122880; dim = 32;  }
  else if (n < 130560) { l = 3; off = 129024; dim = 16;  }
  else                 { l = 4; off = 130560; dim = 8;   }
  int loc = n - off;
  a = loc % 6;
  int hw = loc / 6;
  w = hw % dim; h = hw / dim;
}

// ---------------- pass 0: concat anchors, zero counters ----------------

__global__ __launch_bounds__(256) void prep_kernel(AncPtrs ap,
                                                   float* __restrict__ anchorsAll,
                                                   int* __restrict__ npos) {
  int n = blockIdx.x * 256 + threadIdx.x;
  if (n < N_TOT) {
    int l, loc;
    if      (n <  98304) { l = 0; loc = n; }
    else if (n < 122880) { l = 1; loc = n - 98304; }
    else if (n < 129024) { l = 2; loc = n - 122880; }
    else if (n < 130560) { l = 3; loc = n - 129024; }
    else                 { l = 4; loc = n - 130560; }
#pragma unroll
    for (int k = 0; k < 4; ++k)
      anchorsAll[(size_t)n * 4 + k] = ap.anc[l][(size_t)loc * 4 + k];
  }
  if (blockIdx.x == 0 && threadIdx.x < BQ) npos[threadIdx.x] = 0;
}

// ---------------- pass 1: WMMA distance tiles + per-segment top-9 ----------------
// grid (NSEG, 2 gt-halves, B), 32 threads (1 wave).
// dist^2[m][j] = |gc_m|^2 - 2 gc_m.ac_j + |ac_j|^2  via V_WMMA_F32_16X16X4_F32:
//   A row m = [gcx, gcy, |gc|^2, 0]   B col j = [-2acx, -2acy, 1, 0]   C = |ac_j|^2

__device__ __forceinline__ v8f dist_tile(v2f A, float4 ab, int lane) {
  float acx = 0.5f * (ab.x + ab.z), acy = 0.5f * (ab.y + ab.w);
  float an  = acx * acx + acy * acy;
  v2f Bm;
  Bm.x = (lane < 16) ? (-2.0f * acx) : 1.0f;   // rows K0 / K2
  Bm.y = (lane < 16) ? (-2.0f * acy) : 0.0f;   // rows K1 / K3
  v8f C;
#pragma unroll
  for (int k = 0; k < 8; ++k) C[k] = an;       // C[m][j] = |ac_j|^2
  return __builtin_amdgcn_wmma_f32_16x16x4_f32(
      false, A, false, Bm, (short)0, C, false, false);
}

__device__ __forceinline__ void process_tile(Heap& hp, const float (*lds)[32],
                                             int tileBase, int m, int cb) {
#pragma unroll
  for (int c = 0; c < 8; ++c) {
    int j = cb + c;
    float v = (m < 8) ? lds[m][j] : lds[m - 8][j + 16];
    heap_insert(hp, v, tileBase + j);
  }
}

__global__ __launch_bounds__(32) void topk_seg_kernel(
    const float* __restrict__ anchorsAll, const float* __restrict__ gt_boxes,
    float* __restrict__ candD, int* __restrict__ candI) {
  const int seg = blockIdx.x, g = blockIdx.y, b = blockIdx.z;
  const int lane = threadIdx.x;
  const int r = lane & 15;
  const float4* __restrict__ anc4 = (const float4*)anchorsAll;

  // A matrix (loop invariant): lanes 0-15 hold (K0,K1), lanes 16-31 hold (K2,K3)
  const float* gb = gt_boxes + ((size_t)b * MGT + g * 16 + r) * 4;
  float gx1 = gb[0], gy1 = gb[1], gx2 = gb[2], gy2 = gb[3];
  float gcx = 0.5f * (gx1 + gx2), gcy = 0.5f * (gy1 + gy2);
  float gn  = gcx * gcx + gcy * gcy;
  v2f A;
  A.x = (lane < 16) ? gcx : gn;
  A.y = (lane < 16) ? gcy : 0.0f;

  Heap hp; heap_init(hp);
  __shared__ float ldsD[2][8][32];

  const int t0 = seg * TPS;
  const int t1 = min(t0 + TPS, NUM_TILES);
  const int m  = r;
  const int cb = (lane < 16) ? 0 : 8;

  // 2x unrolled: two independent WMMAs in flight, one barrier pair per pair
  for (int t = t0; t < t1; t += 2) {
    const int ta = t;
    const int tb = min(t + 1, t1 - 1);
    // warm L2/L0 for tiles ~8 ahead (anchor tiles are reused by 16 waves)
    const int tp = min(t + 8, t1 - 1);
    __builtin_prefetch(&anc4[(size_t)tp * 16 + r], 0, 1);

    float4 a0 = anc4[(size_t)ta * 16 + r];
    float4 a1 = anc4[(size_t)tb * 16 + r];

    v8f D0 = dist_tile(A, a0, lane);
    v8f D1 = dist_tile(A, a1, lane);

#pragma unroll
    for (int k = 0; k < 8; ++k) ldsD[0][k][lane] = D0[k];
#pragma unroll
    for (int k = 0; k < 8; ++k) ldsD[1][k][lane] = D1[k];
    __syncthreads();

    process_tile(hp, ldsD[0], ta * 16, m, cb);
    if (t + 1 < t1) process_tile(hp, ldsD[1], tb * 16, m, cb);
    __syncthreads();
  }

  size_t base = ((((size_t)b * 2 + g) * NSEG + seg) * 32 + lane) * TOPK;
#pragma unroll
  for (int k = 0; k < TOPK; ++k) { candD[base + k] = hp.d[k]; candI[base + k] = hp.i[k]; }
}

// ---------------- pass 2: merge candidates -> ATSS threshold per (b, gt) ----------------

__global__ __launch_bounds__(32) void combine_kernel(
    const float* __restrict__ anchorsAll, const float* __restrict__ gt_boxes,
    const float* __restrict__ candD, const int* __restrict__ candI,
    float* __restrict__ thr) {
  const int m = blockIdx.x, b = blockIdx.y, lane = threadIdx.x;
  const int g = m >> 4, r = m & 15;

  Heap hp; heap_init(hp);
  const int total = NSEG * 2 * TOPK;   // 1728 disjoint candidates for this gt
  for (int k = lane; k < total; k += 32) {
    int seg = k / (2 * TOPK);
    int rem = k % (2 * TOPK);
    int laneSel = r + ((rem >= TOPK) ? 16 : 0);
    int slot = rem % TOPK;
    size_t base = ((((size_t)b * 2 + g) * NSEG + seg) * 32 + laneSel) * TOPK + slot;
    heap_insert(hp, candD[base], candI[base]);
  }

  __shared__ float md[32 * TOPK];
  __shared__ int   mi[32 * TOPK];
#pragma unroll
  for (int k = 0; k < TOPK; ++k) { md[lane * TOPK + k] = hp.d[k]; mi[lane * TOPK + k] = hp.i[k]; }
  __syncthreads();

  if (lane == 0) {
    Heap fin; heap_init(fin);
    for (int k = 0; k < 32 * TOPK; ++k) heap_insert(fin, md[k], mi[k]);

    const float* gb = gt_boxes + ((size_t)b * MGT + m) * 4;
    float gx1 = gb[0], gy1 = gb[1], gx2 = gb[2], gy2 = gb[3];
    float gar = (gx2 - gx1) * (gy2 - gy1);
    float s = 0.0f, s2 = 0.0f;
#pragma unroll
    for (int k = 0; k < TOPK; ++k) {
      int idx = fin.i[k];
      float iou = 0.0f;
      if (idx >= 0) {
        float4 ab = ((const float4*)anchorsAll)[idx];
        float aar = (ab.z - ab.x) * (ab.w - ab.y);
        float ix = fmaxf(fminf(ab.z, gx2) - fmaxf(ab.x, gx1), 0.0f);
        float iy = fmaxf(fminf(ab.w, gy2) - fmaxf(ab.y, gy1), 0.0f);
        float inter = ix * iy;
        iou = inter / (aar + gar - inter + EPSF);
      }
      s += iou; s2 += iou * iou;
    }
    float mean = s / 9.0f;
    float var  = fmaxf((s2 - 9.0f * mean * mean) / 8.0f, 0.0f); // ddof=1
    thr[b * MGT + m] = mean + sqrtf(var);
  }
}

// ---------------- pass 3: per-anchor matching ----------------

__global__ __launch_bounds__(256) void match_kernel(
    const float* __restrict__ anchorsAll, const float* __restrict__ gt_boxes,
    const float* __restrict__ thr, int* __restrict__ matched,
    float* __restrict__ miou, int* __restrict__ npos) {
  const int b = blockIdx.y;
  const int n = blockIdx.x * 256 + threadIdx.x;
  __shared__ float gS[MGT * 4];
  __shared__ float tS[MGT];
  if (threadIdx.x < MGT) {
#pragma unroll
    for (int k = 0; k < 4; ++k)
      gS[threadIdx.x * 4 + k] = gt_boxes[((size_t)b * MGT + threadIdx.x) * 4 + k];
    tS[threadIdx.x] = thr[b * MGT + threadIdx.x];
  }
  __syncthreads();

  int mt = -1; float mi_ = 0.0f;
  const bool valid = n < N_TOT;
  if (valid) {
    float4 ab = ((const float4*)anchorsAll)[n];
    float ax1 = ab.x, ay1 = ab.y, ax2 = ab.z, ay2 = ab.w;
    float aar = (ax2 - ax1) * (ay2 - ay1);
    float acx = 0.5f * (ax1 + ax2), acy = 0.5f * (ay1 + ay2);
    for (int m = 0; m < MGT; ++m) {
      float gx1 = gS[m * 4], gy1 = gS[m * 4 + 1], gx2 = gS[m * 4 + 2], gy2 = gS[m * 4 + 3];
      float gar = (gx2 - gx1) * (gy2 - gy1);
      float ix = fmaxf(fminf(ax2, gx2) - fmaxf(ax1, gx1), 0.0f);
      float iy = fmaxf(fminf(ay2, gy2) - fmaxf(ay1, gy1), 0.0f);
      float inter = ix * iy;
      float iou = inter / (aar + gar - inter + EPSF);
      bool inside = (acx >= gx1) && (acx <= gx2) && (acy >= gy1) && (acy <= gy2);
      if (inside && iou >= tS[m]) { mt = m; mi_ = iou; }  // ascending -> max gt id wins
    }
    matched[(size_t)b * N_TOT + n] = mt;
    miou[(size_t)b * N_TOT + n]   = mi_;
  }
  unsigned long long bal = __ballot(valid && (mt >= 0));
  if ((threadIdx.x & 31) == 0 && bal)
    atomicAdd(&npos[b], (int)__popcll(bal));   // integer atomic -> deterministic
}

// ---------------- pass 4: QFL + DFL + GIoU for positive anchors ----------------

__global__ __launch_bounds__(256) void loss_kernel(
    Ptrs p, const float* __restrict__ anchorsAll,
    const float* __restrict__ gt_boxes, const int* __restrict__ gt_labels,
    const int* __restrict__ matched, const float* __restrict__ miou,
    float* __restrict__ partQ, float* __restrict__ partD, float* __restrict__ partG) {
  const int b = blockIdx.y;
  const int n = blockIdx.x * 256 + threadIdx.x;
  float q = 0.0f, dsum = 0.0f, gio = 0.0f;

  if (n < N_TOT) {
    int mt = matched[(size_t)b * N_TOT + n];
    if (mt >= 0) {
      int l, h, w, a, dim;
      decomposeAnchor(n, l, h, w, a, dim);
      const int HW = dim * dim;
      const int sp = h * dim + w;
      const float* clsL = p.cls[l];
      const float* regL = p.reg[l];

      float sc = miou[(size_t)b * N_TOT + n];
      int label = gt_labels[b * MGT + mt];
      const float* gb = gt_boxes + ((size_t)b * MGT + mt) * 4;
      float tx1 = gb[0], ty1 = gb[1], tx2 = gb[2], ty2 = gb[3];

      // ---- QFL (raw layout: channel = a*10 + c) ----
      size_t cbase = ((size_t)b * 60 + a * 10) * HW + sp;
#pragma unroll
      for (int c = 0; c < 10; ++c) {
        float x = clsL[cbase + (size_t)c * HW];
        float sig = 1.0f / (1.0f + __expf(-x));
        if (c == label) {
          float lsp = -softplusf(-x);            // log_sigmoid(x)
          float lsn = -softplusf(x);             // log_sigmoid(-x)
          float bcep = -(sc * lsp + (1.0f - sc) * lsn);
          float dlt = sc - sig;
          q += dlt * dlt * bcep;
        } else {
          q += sig * sig * softplusf(x);         // sig^2 * bce0
        }
      }

      // ---- DFL + decode (channel = a*64 + d4*16 + bin) ----
      float4 abx = ((const float4*)anchorsAll)[n];
      float ax1 = abx.x, ay1 = abx.y, ax2 = abx.z, ay2 = abx.w;
      float aw = ax2 - ax1, ah = ay2 - ay1;
      float enc[4];
      enc[0] = fminf(fmaxf((tx1 - ax1) / aw * 15.0f, 0.0f), 15.0f);
      enc[1] = fminf(fmaxf((ty1 - ay1) / ah * 15.0f, 0.0f), 15.0f);
      enc[2] = fminf(fmaxf((tx2 - ax2) / aw * 15.0f, 0.0f), 15.0f);
      enc[3] = fminf(fmaxf((ty2 - ay2) / ah * 15.0f, 0.0f), 15.0f);

      float dist[4];
      size_t rbase = ((size_t)b * 384 + a * 64) * HW + sp;
#pragma unroll
      for (int d4 = 0; d4 < 4; ++d4) {
        float x[16];
        float mx = -3.0e38f;
#pragma unroll
        for (int t = 0; t < 16; ++t) {
          x[t] = regL[rbase + (size_t)(d4 * 16 + t) * HW];
          mx = fmaxf(mx, x[t]);
        }
        float se = 0.0f, ex = 0.0f;
#pragma unroll
        for (int t = 0; t < 16; ++t) {
          float e = __expf(x[t] - mx);
          se += e; ex += e * (float)t;
        }
        float logZ = mx + __logf(se);
        float e = enc[d4];
        int dl = min(max((int)floorf(e), 0), 15);
        int dr = min(dl + 1, 15);
        float wl = (float)(dl + 1) - e;
        float wr = e - (float)dl;
        float xl = 0.0f, xr = 0.0f;
#pragma unroll
        for (int t = 0; t < 16; ++t) { if (t == dl) xl = x[t]; if (t == dr) xr = x[t]; }
        dsum += (logZ - xl) * wl + (logZ - xr) * wr;
        dist[d4] = (ex / se) * (1.0f / 15.0f);
      }

      // ---- GIoU ----
      float px1 = ax1 - dist[0] * aw;
      float py1 = ay1 - dist[1] * ah;
      float px2 = ax2 + dist[2] * aw;
      float py2 = ay2 + dist[3] * ah;
      float ix = fmaxf(fminf(px2, tx2) - fmaxf(px1, tx1), 0.0f);
      float iy = fmaxf(fminf(py2, ty2) - fmaxf(py1, ty1), 0.0f);
      float inter = ix * iy;
      float par = (px2 - px1) * (py2 - py1);
      float tar = (tx2 - tx1) * (ty2 - ty1);
      float uni = par + tar - inter + EPSF;
      float iou = inter / uni;
      float ew = fmaxf(fmaxf(px2, tx2) - fminf(px1, tx1), 0.0f);
      float eh = fmaxf(fmaxf(py2, ty2) - fminf(py1, ty1), 0.0f);
      float ear = ew * eh + EPSF;
      gio += 1.0f - (iou - (ear - uni) / ear);
    }
  }

  // deterministic fixed-order block reduction
  __shared__ float rQ[256], rD[256], rG[256];
  int tid = threadIdx.x;
  rQ[tid] = q; rD[tid] = dsum; rG[tid] = gio;
  __syncthreads();
  for (int s = 128; s > 0; s >>= 1) {
    if (tid < s) { rQ[tid] += rQ[tid + s]; rD[tid] += rD[tid + s]; rG[tid] += rG[tid + s]; }
    __syncthreads();
  }
  if (tid == 0) {
    int pb = b * gridDim.x + blockIdx.x;
    partQ[pb] = rQ[0]; partD[pb] = rD[0]; partG[pb] = rG[0];
  }
}

// ---------------- pass 5: final deterministic reduction ----------------

__global__ __launch_bounds__(32) void final_kernel(
    const float* __restrict__ partQ, const float* __restrict__ partD,
    const float* __restrict__ partG, const int* __restrict__ npos,
    int nblocks, float* __restrict__ out) {
  int lane = threadIdx.x;
  __shared__ float sQ[BQ], sD[BQ], sG[BQ];
  __shared__ int sH[BQ];
  if (lane < BQ) {
    float qq = 0.0f, dd = 0.0f, gg = 0.0f;
    for (int k = 0; k < nblocks; ++k) {
      qq += partQ[lane * nblocks + k];
      dd += partD[lane * nblocks + k];
      gg += partG[lane * nblocks + k];
    }
    int np = npos[lane];
    float den = (float)max(np, 1);
    bool has = np > 0;
    sQ[lane] = has ? qq / den : 0.0f;
    sD[lane] = has ? dd / (den * 4.0f) : 0.0f;
    sG[lane] = has ? gg / den : 0.0f;
    sH[lane] = has ? 1 : 0;
  }
  __syncthreads();
  if (lane == 0) {
    float tq = 0.0f, td = 0.0f, tg = 0.0f; int hv = 0;
    for (int k = 0; k < BQ; ++k) { tq += sQ[k]; td += sD[k]; tg += sG[k]; hv += sH[k]; }
    float valid = (float)max(hv, 1);
    tq /= valid; td /= valid; tg /= valid;
    out[0] = tq; out[1] = td; out[2] = tg; out[3] = tq + td + tg;
  }
}

// ---------------- launcher ----------------

extern "C" void kernel_launch(void* const* d_in, const int* in_sizes, int n_in,
                              void* d_out, int out_size, void* d_ws, size_t ws_size,
                              hipStream_t stream) {
  // setup_inputs() dict order is interleaved: cls_l, reg_l, anchors_l per level.
  Ptrs p; AncPtrs ap;
  for (int l = 0; l < 5; ++l) {
    p.cls[l]  = (const float*)d_in[3 * l + 0];
    p.reg[l]  = (const float*)d_in[3 * l + 1];
    ap.anc[l] = (const float*)d_in[3 * l + 2];
  }
  const float* gt_boxes  = (const float*)d_in[15];
  const int*   gt_labels = (const int*)d_in[16];

  char* ws = (char*)d_ws;
  size_t off = 0;
  auto take = [&](size_t bytes) -> char* {
    char* r = ws + off;
    off = (off + bytes + 255) & ~(size_t)255;
    return r;
  };
  float* anchorsAll = (float*)take((size_t)N_TOT * 4 * sizeof(float));
  float* thr        = (float*)take((size_t)BQ * MGT * sizeof(float));
  int*   matched    = (int*)  take((size_t)BQ * N_TOT * sizeof(int));
  float* miou       = (float*)take((size_t)BQ * N_TOT * sizeof(float));
  size_t candCount  = (size_t)BQ * 2 * NSEG * 32 * TOPK;
  float* candD      = (float*)take(candCount * sizeof(float));
  int*   candI      = (int*)  take(candCount * sizeof(int));
  int*   npos       = (int*)  take((size_t)BQ * sizeof(int));
  float* partQ      = (float*)take((size_t)BQ * NBLK * sizeof(float));
  float* partD      = (float*)take((size_t)BQ * NBLK * sizeof(float));
  float* partG      = (float*)take((size_t)BQ * NBLK * sizeof(float));

  prep_kernel<<<NBLK, 256, 0, stream>>>(ap, anchorsAll, npos);
  topk_seg_kernel<<<dim3(NSEG, 2, BQ), 32, 0, stream>>>(anchorsAll, gt_boxes, candD, candI);
  combine_kernel<<<dim3(MGT, BQ), 32, 0, stream>>>(anchorsAll, gt_boxes, candD, candI, thr);
  match_kernel<<<dim3(NBLK, BQ), 256, 0, stream>>>(anchorsAll, gt_boxes, thr, matched, miou, npos);
  loss_kernel<<<dim3(NBLK, BQ), 256, 0, stream>>>(p, anchorsAll, gt_boxes, gt_labels,
                                                  matched, miou, partQ, partD, partG);
  final_kernel<<<1, 32, 0, stream>>>(partQ, partD, partG, npos, NBLK, (float*)d_out);
}